// Warp_9182640079483
// MI455X (gfx1250) — compile-verified
//
#include <hip/hip_runtime.h>

// Bilinear flow warp for B=8, C=64, H=256, W=448 (fp32), memory-bound gather.
// Per pixel: weights computed once (edge-clamp permutation folded into the
// weights), then 64 channels of {2x global_load_b64 pair-gather, 4 VALU,
// 1 non-temporal store}. Output streamed NT so the 235MB of writes don't
// evict the input working set from the 192MB L2.

#define B_  8
#define C_  64
#define H_  256
#define W_  448
#define HW_ (H_ * W_)

// <2 x float> with 4-byte alignment: lets the backend emit global_load_b64
// at float-aligned addresses (gfx1250 handles 4B-aligned b64 in DWORD mode).
typedef float v2f __attribute__((ext_vector_type(2)));
typedef v2f   v2f_a4 __attribute__((aligned(4)));

__global__ __launch_bounds__(W_) void warp_bilinear_kernel(
    const float* __restrict__ img,   // [B, C, H, W]
    const float* __restrict__ flow,  // [B, 2, H, W] (dx, dy planes)
    float* __restrict__ out)         // [B, C, H, W]
{
    const int x = threadIdx.x;   // 0..447  (one block = one row; 14 wave32s)
    const int y = blockIdx.x;    // 0..255
    const int b = blockIdx.y;    // 0..7

    const int pix = y * W_ + x;

    // Flow planes are separate [H,W] slabs -> two perfectly coalesced loads.
    const size_t fbase = (size_t)b * 2 * HW_ + pix;
    const float dx = flow[fbase];
    const float dy = flow[fbase + HW_];

    // Sample position and bilinear weights (match reference math).
    const float sx = (float)x + dx;
    const float sy = (float)y + dy;
    const float x0f = floorf(sx);
    const float y0f = floorf(sy);
    const float x1f = x0f + 1.0f;
    const float y1f = y0f + 1.0f;

    const float wx1 = sx - x0f;
    const float wx0 = 1.0f - wx1;
    const float wy1 = sy - y0f;
    const float wy0 = 1.0f - wy1;

    // Out-of-bounds contributes zero: fold validity into the weights.
    const float vx0 = (x0f >= 0.0f && x0f <= (float)(W_ - 1)) ? 1.0f : 0.0f;
    const float vx1 = (x1f >= 0.0f && x1f <= (float)(W_ - 1)) ? 1.0f : 0.0f;
    const float vy0 = (y0f >= 0.0f && y0f <= (float)(H_ - 1)) ? 1.0f : 0.0f;
    const float vy1 = (y1f >= 0.0f && y1f <= (float)(H_ - 1)) ? 1.0f : 0.0f;

    const float w00 = wy0 * wx0 * (vy0 * vx0);  // (y0,x0)
    const float w01 = wy0 * wx1 * (vy0 * vx1);  // (y0,x1)
    const float w10 = wy1 * wx0 * (vy1 * vx0);  // (y1,x0)
    const float w11 = wy1 * wx1 * (vy1 * vx1);  // (y1,x1)

    // Clamped integer gather coords.
    const int xi0 = (int)fminf(fmaxf(x0f, 0.0f), (float)(W_ - 1));
    const int xi1 = (int)fminf(fmaxf(x1f, 0.0f), (float)(W_ - 1));
    const int yi0 = (int)fminf(fmaxf(y0f, 0.0f), (float)(H_ - 1));
    const int yi1 = (int)fminf(fmaxf(y1f, 0.0f), (float)(H_ - 1));

    // Pair-gather base: b64 at xb covers columns (xb, xb+1), always in-bounds.
    // t/u in {0,1} say which half of the pair xi0/xi1 actually live in;
    // fold that permutation into the weights ONCE instead of per-channel
    // selects:  w00*pair[t] + w01*pair[u] == wA0*pair.x + wB0*pair.y.
    const int xb = (xi0 < (W_ - 2)) ? xi0 : (W_ - 2);
    const int t  = xi0 - xb;   // 1 only on right clamp
    const int u  = xi1 - xb;   // 0 only on left clamp

    const float wA0 = (t == 0 ? w00 : 0.0f) + (u == 0 ? w01 : 0.0f);
    const float wB0 = (t != 0 ? w00 : 0.0f) + (u != 0 ? w01 : 0.0f);
    const float wA1 = (t == 0 ? w10 : 0.0f) + (u == 0 ? w11 : 0.0f);
    const float wB1 = (t != 0 ? w10 : 0.0f) + (u != 0 ? w11 : 0.0f);

    const int o0 = yi0 * W_ + xb;   // row y0 pair offset (elements)
    const int o1 = yi1 * W_ + xb;   // row y1 pair offset (elements)

    const float* __restrict__ p = img + (size_t)b * C_ * HW_;
    float* __restrict__       q = out + (size_t)b * C_ * HW_ + pix;

    // Warm the first gather cachelines (gfx1250 global_prefetch_b8) while
    // the weight VALU work drains.
    __builtin_prefetch(p + o0, 0, 3);
    __builtin_prefetch(p + o1, 0, 3);

    // 64 channels: 2 pair-gathers (b64) + 1 mul + 3 FMA + 1 NT store each.
    // Unroll-by-8 -> clauses of 16 outstanding b64 loads per thread.
#pragma unroll 8
    for (int c = 0; c < C_; ++c) {
        const v2f r0 = *(const v2f_a4*)(p + o0);  // columns (xb, xb+1) @ y0
        const v2f r1 = *(const v2f_a4*)(p + o1);  // columns (xb, xb+1) @ y1
        const float r = fmaf(wB1, r1.y,
                        fmaf(wA1, r1.x,
                        fmaf(wB0, r0.y, wA0 * r0.x)));
        __builtin_nontemporal_store(r, q);  // stream: don't pollute L2
        p += HW_;
        q += HW_;
    }
}

extern "C" void kernel_launch(void* const* d_in, const int* in_sizes, int n_in,
                              void* d_out, int out_size, void* d_ws, size_t ws_size,
                              hipStream_t stream) {
    (void)in_sizes; (void)n_in; (void)ws_size; (void)d_ws; (void)out_size;
    const float* img  = (const float*)d_in[0];  // [8,64,256,448] fp32
    const float* flow = (const float*)d_in[1];  // [8,2,256,448]  fp32
    float*       out  = (float*)d_out;          // [8,64,256,448] fp32

    dim3 grid(H_, B_);   // one block per (row, batch)
    dim3 block(W_);      // 448 threads = 14 wave32s = one full row
    warp_bilinear_kernel<<<grid, block, 0, stream>>>(img, flow, out);
}